// HyenaSE_31653908972217
// MI455X (gfx1250) — compile-verified
//
#include <hip/hip_runtime.h>

// CDNA5 / gfx1250 Hyena-SE gated causal depthwise conv via fp32 WMMA.
//
//   out[b,d,l] = q[b,d,l] * sum_{u=0..127} k[b,d,l-127+u]*x[b,d,l-127+u]*h[d%G, u]
//
// One wave32 processes one (row, 1024-output segment). Each 256-output tile is
// a 16x16 banded-Toeplitz matmul: Y[r,i] += sum_n A_n(16x16 shifted kx chunks)
// x B_n(banded filter), done as 9x4 V_WMMA_F32_16X16X4_F32 (full fp32).
// B fragments are register-resident and reused across the 4 tiles; the filter
// sits in zero-padded LDS so B loads need no range guards (no exec masking).

typedef __attribute__((ext_vector_type(2))) float v2f;
typedef __attribute__((ext_vector_type(4))) float v4f;
typedef __attribute__((ext_vector_type(8))) float v8f;

namespace {
constexpr int kL  = 4096;              // sequence length
constexpr int kG  = 256;               // filter groups (filter index = d % G)
constexpr int kHL = 128;               // filter taps
constexpr int kTile = 256;             // outputs per 16x16 WMMA tile
constexpr int kTilesPerSeg = 4;        // tiles per wave
constexpr int kSeg = kTile * kTilesPerSeg;   // 1024 outputs per wave
constexpr int kStage = kSeg + kHL;     // staged kx window incl. halo = 1152
constexpr int kHPad = 16 + kHL + 16;   // zero-padded filter = 160
constexpr int kWaves = 8;              // waves per block
}

__global__ __launch_bounds__(kWaves * 32)
void hyena_se_wmma_kernel(const float* __restrict__ x,
                          const float* __restrict__ k,
                          const float* __restrict__ q,
                          const float* __restrict__ h,
                          float* __restrict__ out) {
    __shared__ float lds_kx[kWaves][kStage];   // per-wave kx window
    __shared__ float lds_h [kWaves][kHPad];    // per-wave zero-padded filter

    const int lane = threadIdx.x & 31;
    const int wv   = threadIdx.x >> 5;

    const int segs_per_row = kL / kSeg;                   // 4
    const int task = blockIdx.x * kWaves + wv;
    const int row  = task / segs_per_row;                 // b*D + d
    const int seg  = task % segs_per_row;
    const int s0   = seg * kSeg;                          // first output of segment
    const size_t rb = (size_t)row * kL;
    const int g = row & (kG - 1);                         // d % 256 (D % G == 0)

    // ---- Stage filter into zero-padded LDS: [0,16) and [144,160) stay zero.
    #pragma unroll
    for (int it = 0; it < kHPad / 32; ++it)               // 5 iters
        lds_h[wv][it * 32 + lane] = 0.0f;
    __builtin_amdgcn_wave_barrier();
    *(v4f*)&lds_h[wv][16 + lane * 4] = *(const v4f*)(h + (size_t)g * kHL + lane * 4);

    // ---- Stage kx = k*x for [s0-128, s0+1024): 9 float4 per lane, coalesced.
    #pragma unroll
    for (int it = 0; it < kStage / 128; ++it) {           // 9 iters
        const int f4 = it * 32 + lane;                    // 0..287
        const int p  = s0 - kHL + f4 * 4;
        v4f kv = {}, xv = {};
        if (p >= 0) {                                     // causal left pad -> zeros
            kv = *(const v4f*)(k + rb + p);
            xv = *(const v4f*)(x + rb + p);
        }
        *(v4f*)&lds_kx[wv][f4 * 4] = kv * xv;
    }
    __builtin_amdgcn_wave_barrier();

    const int half = lane >> 4;   // K lane-half for A/B fragments
    const int i    = lane & 15;   // A row (output chunk) == B/D column

    // ---- Register-resident B fragments: B_n[j,i] = h[16n + j - i - 1] (padded).
    v2f bfrag[9][4];
    #pragma unroll
    for (int n = 0; n < 9; ++n) {
        #pragma unroll
        for (int kk = 0; kk < 4; ++kk) {
            const int ux = 16 + 16 * n + 4 * kk + 2 * half - i - 1;  // in [0,158]
            bfrag[n][kk].x = lds_h[wv][ux];
            bfrag[n][kk].y = lds_h[wv][ux + 1];
        }
    }

    // ---- 4 tiles of 256 outputs, reusing staged kx + resident B.
    #pragma unroll
    for (int tt = 0; tt < kTilesPerSeg; ++tt) {
        v8f acc = {};
        #pragma unroll
        for (int n = 0; n < 9; ++n) {
            #pragma unroll
            for (int kk = 0; kk < 4; ++kk) {
                // A[r=i][j0..j0+1]; staged index = 256*tt + 16*(r+n) + j
                v2f a = *(const v2f*)&lds_kx[wv][tt * kTile + 16 * (i + n)
                                                + 4 * kk + 2 * half];
                acc = __builtin_amdgcn_wmma_f32_16x16x4_f32(
                    /*neg_a=*/false, a, /*neg_b=*/false, bfrag[n][kk],
                    /*c_mod=*/(short)0, acc, /*reuse_a=*/false, /*reuse_b=*/false);
            }
        }
        // Epilogue: gate with q and store. acc[p] holds Y[M=p+8*half][N=i].
        const int l0 = s0 + tt * kTile;
        #pragma unroll
        for (int p = 0; p < 8; ++p) {
            const int M = p + 8 * half;
            const size_t pos = rb + (size_t)(l0 + 16 * M + i);
            out[pos] = q[pos] * acc[p];
        }
    }
}

extern "C" void kernel_launch(void* const* d_in, const int* in_sizes, int n_in,
                              void* d_out, int out_size, void* d_ws, size_t ws_size,
                              hipStream_t stream) {
    const float* x = (const float*)d_in[0];
    const float* k = (const float*)d_in[1];
    const float* q = (const float*)d_in[2];
    const float* h = (const float*)d_in[3];
    float* out = (float*)d_out;

    const int total = in_sizes[0];                 // B*D*L
    const int rows  = total / kL;                  // B*D = 8192
    const int tasks = rows * (kL / kSeg);          // 32768 wave tasks
    const int blocks = tasks / kWaves;             // 4096

    hyena_se_wmma_kernel<<<blocks, kWaves * 32, 0, stream>>>(x, k, q, h, out);
}